// KDEHyperGraphConv_38027640438848
// MI455X (gfx1250) — compile-verified
//
#include <hip/hip_runtime.h>
#include <cstddef>

#define Bb 16
#define Mm 2048
#define Cc 128

typedef __attribute__((ext_vector_type(2))) float v2f;
typedef __attribute__((ext_vector_type(8))) float v8f;

__device__ __forceinline__ v8f wmma_f32_k4(v2f a, v2f b, v8f c) {
  // D = A(16x4,f32) * B(4x16,f32) + C(16x16,f32)
  return __builtin_amdgcn_wmma_f32_16x16x4_f32(false, a, false, b, (short)0, c,
                                               false, false);
}

__device__ __forceinline__ float wave_reduce_add(float v) {
#pragma unroll
  for (int off = 1; off < 32; off <<= 1) v += __shfl_xor(v, off, 32);
  return v;
}

// ---------------- K1: row squared norms (+ zero hsum) ----------------
__global__ __launch_bounds__(256) void sq_kernel(const float* __restrict__ x,
                                                 float* __restrict__ sq,
                                                 float* __restrict__ hsum) {
  const int t = threadIdx.x, w = t >> 5, lane = t & 31;
  const int row = blockIdx.x * 8 + w;  // 4096 blocks * 8 rows = 32768 rows
  const float4 v = *(const float4*)(x + (size_t)row * Cc + lane * 4);
  float s = v.x * v.x + v.y * v.y + v.z * v.z + v.w * v.w;
  s = wave_reduce_add(s);
  if (lane == 0) sq[row] = s;
  if (blockIdx.x == 0 && t < Bb) hsum[t] = 0.f;  // re-zeroed every launch
}

// ---------------- K2/K4: Gram tiles via WMMA f32 16x16x4 ----------------
// LDS tile layout: c-pairs adjacent so each B operand is ONE ds_load_b64
// straight into the WMMA operand pair. word(c,r) = (c>>1)*32 +
// 2*((r + (c>>2? no: pair>>1)) & 15) + (c&1); rotation keeps reads
// conflict-free and cuts staging-write conflicts to 2-way.
// mode 0: sum of off-diagonal dist -> atomicAdd(hsum[b])
// mode 1: rho[b,i] = sum_j exp(-clip(d2)*inv2h2[b])
__global__ __launch_bounds__(256) void gram_kernel(
    const float* __restrict__ x, const float* __restrict__ sq,
    const float* __restrict__ inv2h2, float* __restrict__ hsum,
    float* __restrict__ rho, int mode) {
  __shared__ float Bst[2][2048];  // 64 c-pairs x 16 rows x 2
  __shared__ float sqj_s[2][16];
  __shared__ float sqi_s[8][16];
  __shared__ float red[8];

  const int t = threadIdx.x;
  const int w = t >> 5;
  const int lane = t & 31;
  const int g = lane >> 4;  // half-wave group
  const int ln = lane & 15;
  const int b = blockIdx.y;
  const int i0 = blockIdx.x * 128 + w * 16;
  const float* xb = x + (size_t)b * Mm * Cc;

  // Preload A rows: lane holds row i0+ln, K pattern k = 4q + 2g (+1)
  v2f areg[32];
  {
    const float* rowp = xb + (size_t)(i0 + ln) * Cc + 2 * g;
#pragma unroll
    for (int q = 0; q < 32; ++q) areg[q] = *(const v2f*)(rowp + 4 * q);
  }
  if (lane < 16) sqi_s[w][lane] = sq[b * Mm + i0 + lane];
  const float inv = (mode == 1) ? inv2h2[b] : 0.f;

  float rowacc[8];
#pragma unroll
  for (int r8 = 0; r8 < 8; ++r8) rowacc[r8] = 0.f;
  float totacc = 0.f;

  const int r = t >> 4;          // staging row 0..15
  const int c8 = (t & 15) << 3;  // staging col group

  // pair-layout store of 8 consecutive c values for row r
  auto stage_tile = [&](float* dst, const float4& v0, const float4& v1) {
    v2f ps[4] = {{v0.x, v0.y}, {v0.z, v0.w}, {v1.x, v1.y}, {v1.z, v1.w}};
#pragma unroll
    for (int s = 0; s < 4; ++s) {
      const int pr = (c8 >> 1) + s;  // c-pair index 0..63
      const int wd = pr * 32 + 2 * ((r + (pr >> 1)) & 15);
      *(v2f*)&dst[wd] = ps[s];
    }
  };

  {  // prologue: stage j-tile 0 into buffer 0
    const float* src = xb + (size_t)r * Cc + c8;
    stage_tile(&Bst[0][0], *(const float4*)(src), *(const float4*)(src + 4));
    if (t < 16) sqj_s[0][t] = sq[b * Mm + t];
  }

  float4 nv0 = {0, 0, 0, 0}, nv1 = {0, 0, 0, 0};
  float nsq = 0.f;
  for (int jt = 0; jt < Mm / 16; ++jt) {
    const int j0 = jt * 16;
    const int p = jt & 1;
    const bool havenext = (jt + 1 < Mm / 16);
    if (havenext) {  // issue next tile's global loads now; drain after compute
      const float* src = xb + (size_t)(j0 + 16 + r) * Cc + c8;
      nv0 = *(const float4*)(src);
      nv1 = *(const float4*)(src + 4);
      if (t < 16) nsq = sq[b * Mm + j0 + 16 + t];
      if (jt + 2 < Mm / 16)
        __builtin_prefetch(xb + (size_t)(j0 + 32 + r) * Cc + c8, 0, 3);
    }
    __syncthreads();  // buf p visible; prior reads of buf p^1 complete

    const float* bp = &Bst[p][0];
    v8f acc = {0.f, 0.f, 0.f, 0.f, 0.f, 0.f, 0.f, 0.f};
#pragma unroll
    for (int q = 0; q < 32; ++q) {
      const int p2 = 2 * q + g;  // c-pair of this operand, (p2>>1)==q
      const v2f bb = *(const v2f*)&bp[p2 * 32 + 2 * ((ln + q) & 15)];
      acc = wmma_f32_k4(areg[q], bb, acc);
    }

    const float sqjv = sqj_s[p][ln];
    if (mode == 0) {
#pragma unroll
      for (int r8 = 0; r8 < 8; ++r8) {
        const int m = r8 + 8 * g;  // C layout row
        float d2 = sqi_s[w][m] + sqjv - 2.f * acc[r8];
        float dist = sqrtf(fmaxf(d2, 1e-12f));
        if (i0 + m != j0 + ln) totacc += dist;
      }
    } else {
#pragma unroll
      for (int r8 = 0; r8 < 8; ++r8) {
        const int m = r8 + 8 * g;
        float d2 = sqi_s[w][m] + sqjv - 2.f * acc[r8];
        rowacc[r8] += expf(-fmaxf(d2, 1e-12f) * inv);
      }
    }

    if (havenext) {  // drain staged registers into the alternate buffer
      stage_tile(&Bst[p ^ 1][0], nv0, nv1);
      if (t < 16) sqj_s[p ^ 1][t] = nsq;
    }
  }

  if (mode == 0) {
    float s = wave_reduce_add(totacc);
    if (lane == 0) red[w] = s;
    __syncthreads();
    if (t == 0) {
      float tot = 0.f;
#pragma unroll
      for (int i = 0; i < 8; ++i) tot += red[i];
      atomicAdd(&hsum[b], tot);
    }
  } else {
#pragma unroll
    for (int r8 = 0; r8 < 8; ++r8) {
      float s = rowacc[r8];
#pragma unroll
      for (int off = 1; off < 16; off <<= 1) s += __shfl_xor(s, off, 16);
      if (ln == 0) rho[b * Mm + i0 + r8 + 8 * g] = s;
    }
  }
}

// ---------------- K3: h, 1/(2h^2), prefix table P ----------------
__global__ void finalize_kernel(const float* __restrict__ hsum,
                                float* __restrict__ inv2h2,
                                float* __restrict__ Ptab) {
  const int t = threadIdx.x;
  if (t < Bb) {
    float hv = hsum[t] / (float)(Mm * (Mm - 1));
    float h = fmaxf(hv, 1e-6f);
    inv2h2[t] = 1.f / (2.f * h * h);
  }
  if (t == 0) {  // P(r) = sum_{u<r} 1/(M-u), r = 0..M
    float acc = 0.f;
    Ptab[0] = 0.f;
    for (int r = 0; r < Mm; ++r) {
      acc += 1.f / (float)(Mm - r);
      Ptab[r + 1] = acc;
    }
  }
}

// ---------------- K5: ranks by counting ----------------
__global__ __launch_bounds__(256) void rank_kernel(const float* __restrict__ rho,
                                                   int* __restrict__ rnk) {
  __shared__ float rs[Mm];
  const int b = blockIdx.x, t = threadIdx.x;
  for (int i = t; i < Mm; i += 256) rs[i] = rho[b * Mm + i];
  __syncthreads();
  for (int i = t; i < Mm; i += 256) {
    const float ri = rs[i];
    int c = 0;
    for (int j = 0; j < Mm; ++j) c += (rs[j] < ri) ? 1 : 0;
    rnk[b * Mm + i] = c;
  }
}

// ---------------- K6: y = Dv^-1/2 * A * Dv^-1/2 * x  (A synthesized) ----
// LDS x-tile layout: k-pairs adjacent -> one ds_load_b64 per B operand.
// word(k,c) = (k>>1)*256 + 2*((c + 16*((k>>1)&1)) & 127) + (k&1);
// the +16 parity shift puts the g=1 half exactly 32 banks away -> no
// read conflicts.
__global__ __launch_bounds__(256) void prop_kernel(
    const float* __restrict__ x, const int* __restrict__ rnk,
    const float* __restrict__ Ptab, float* __restrict__ y) {
  __shared__ float Xs[2][2048];  // 8 k-pairs x 128 cols x 2
  __shared__ float Ps[Mm + 1];
  __shared__ int rks[2][16];
  __shared__ float dvk[2][16];
  __shared__ float dvi_s[8][16];

  const int t = threadIdx.x;
  const int w = t >> 5, lane = t & 31, g = lane >> 4, ln = lane & 15;
  const int b = blockIdx.y;
  const int i0 = blockIdx.x * 128 + w * 16;
  const float* xb = x + (size_t)b * Mm * Cc;

  for (int idx = t; idx <= Mm; idx += 256) Ps[idx] = Ptab[idx];
  const int rm = rnk[b * Mm + i0 + ln];  // A-operand row rank (M = ln)
  if (lane < 16)
    dvi_s[w][lane] = rsqrtf((float)(rnk[b * Mm + i0 + lane] + 1));

  const v8f z = {0.f, 0.f, 0.f, 0.f, 0.f, 0.f, 0.f, 0.f};
  v8f acc[8];
#pragma unroll
  for (int ct = 0; ct < 8; ++ct) acc[ct] = z;

  const int r = t >> 4;          // staging k-row 0..15
  const int c8 = (t & 15) << 3;  // staging col group
  const int pkw = r >> 1;        // staging k-pair
  const int parw = r & 1;
  const int sww = 16 * (pkw & 1);

  auto stage_tile = [&](float* dst, const float4& v0, const float4& v1) {
    float vv[8] = {v0.x, v0.y, v0.z, v0.w, v1.x, v1.y, v1.z, v1.w};
#pragma unroll
    for (int u = 0; u < 8; ++u) {
      const int cs = (c8 + u + sww) & 127;
      dst[pkw * 256 + 2 * cs + parw] = vv[u];
    }
  };

  {  // prologue: stage k-tile 0 into buffer 0
    const float* src = xb + (size_t)r * Cc + c8;
    stage_tile(&Xs[0][0], *(const float4*)src, *(const float4*)(src + 4));
    if (t < 16) {
      const int rv = rnk[b * Mm + t];
      rks[0][t] = rv;
      dvk[0][t] = rsqrtf((float)(rv + 1));
    }
  }

  float4 nv0 = {0, 0, 0, 0}, nv1 = {0, 0, 0, 0};
  int nrk = 0;
  for (int kt = 0; kt < Mm / 16; ++kt) {
    const int k0 = kt * 16;
    const int p = kt & 1;
    const bool havenext = (kt + 1 < Mm / 16);
    if (havenext) {  // issue next tile's global loads before compute
      const float* src = xb + (size_t)(k0 + 16 + r) * Cc + c8;
      nv0 = *(const float4*)src;
      nv1 = *(const float4*)(src + 4);
      if (t < 16) nrk = rnk[b * Mm + k0 + 16 + t];
      if (kt + 2 < Mm / 16)
        __builtin_prefetch(xb + (size_t)(k0 + 32 + r) * Cc + c8, 0, 3);
    }
    __syncthreads();

    // synthesize the A-operand: A(m,k) = P(min(r_m, r_k)+1) * dvs_k
    v2f am[4];
#pragma unroll
    for (int q = 0; q < 4; ++q) {
      const int kl = 4 * q + 2 * g;
      const int r0 = rks[p][kl], r1 = rks[p][kl + 1];
      const int m0 = (rm < r0 ? rm : r0) + 1;
      const int m1 = (rm < r1 ? rm : r1) + 1;
      am[q].x = Ps[m0] * dvk[p][kl];
      am[q].y = Ps[m1] * dvk[p][kl + 1];
    }
    const float* xp = &Xs[p][0];
#pragma unroll
    for (int ct = 0; ct < 8; ++ct) {
#pragma unroll
      for (int q = 0; q < 4; ++q) {
        const int pk = 2 * q + g;  // k-pair of this operand (pk&1 == g)
        const int cs = (16 * ct + ln + 16 * g) & 127;
        const v2f bb = *(const v2f*)&xp[pk * 256 + 2 * cs];
        acc[ct] = wmma_f32_k4(am[q], bb, acc[ct]);
      }
    }

    if (havenext) {  // drain into alternate buffer
      stage_tile(&Xs[p ^ 1][0], nv0, nv1);
      if (t < 16) {
        rks[p ^ 1][t] = nrk;
        dvk[p ^ 1][t] = rsqrtf((float)(nrk + 1));
      }
    }
  }

#pragma unroll
  for (int ct = 0; ct < 8; ++ct) {
#pragma unroll
    for (int r8 = 0; r8 < 8; ++r8) {
      const int m = r8 + 8 * g;
      y[((size_t)b * Mm + i0 + m) * Cc + 16 * ct + ln] =
          dvi_s[w][m] * acc[ct][r8];
    }
  }
}

// ---------------- K7: out = SiLU(y @ W^T), in-place on d_out ----------------
// W^T half staged as c-pairs adjacent: word(c,o) = (c>>1)*128 +
// 2*((o + 16*((c>>1)&1)) & 63) + (c&1) -> one ds_load_b64 per operand.
__global__ __launch_bounds__(256) void outproj_kernel(const float* __restrict__ Wm,
                                                      float* yout) {
  __shared__ float Wt[64 * 128];  // 64 c-pairs x 64 o x 2 (32KB per half)
  const int t = threadIdx.x;
  const int w = t >> 5, lane = t & 31, g = lane >> 4, ln = lane & 15;
  const int row0 = blockIdx.x * 128 + w * 16;  // 256 blocks cover 32768 rows

  // Preload this wave's 16 y-rows into registers BEFORE any overwrite.
  v2f areg[32];
  {
    const float* rowp = yout + (size_t)(row0 + ln) * Cc + 2 * g;
#pragma unroll
    for (int q = 0; q < 32; ++q) areg[q] = *(const v2f*)(rowp + 4 * q);
  }

  const v8f z = {0.f, 0.f, 0.f, 0.f, 0.f, 0.f, 0.f, 0.f};
  v8f acc[8];
#pragma unroll
  for (int ct = 0; ct < 8; ++ct) acc[ct] = z;

#pragma unroll
  for (int h = 0; h < 2; ++h) {  // two 64-col halves of W^T in LDS
    __syncthreads();
    for (int idx = t; idx < 64 * Cc; idx += 256) {
      const int ol = idx >> 7;       // 0..63
      const int c = idx & (Cc - 1);  // 0..127
      const int pc = c >> 1;
      const int os = (ol + 16 * (pc & 1)) & 63;
      Wt[pc * 128 + 2 * os + (c & 1)] = Wm[(64 * h + ol) * Cc + c];
    }
    __syncthreads();
#pragma unroll
    for (int ctl = 0; ctl < 4; ++ctl) {
      const int ct = 4 * h + ctl;
#pragma unroll
      for (int q = 0; q < 32; ++q) {
        const int pc = 2 * q + g;  // c-pair (pc&1 == g)
        const int os = (16 * ctl + ln + 16 * g) & 63;
        const v2f bb = *(const v2f*)&Wt[pc * 128 + 2 * os];
        acc[ct] = wmma_f32_k4(areg[q], bb, acc[ct]);
      }
    }
  }

#pragma unroll
  for (int ct = 0; ct < 8; ++ct) {
#pragma unroll
    for (int r8 = 0; r8 < 8; ++r8) {
      const int m = r8 + 8 * g;
      const float v = acc[ct][r8];
      yout[(size_t)(row0 + m) * Cc + 16 * ct + ln] = v / (1.f + expf(-v));
    }
  }
}

// ---------------- host-side launch ----------------
extern "C" void kernel_launch(void* const* d_in, const int* in_sizes, int n_in,
                              void* d_out, int out_size, void* d_ws,
                              size_t ws_size, hipStream_t stream) {
  (void)in_sizes; (void)n_in; (void)out_size; (void)ws_size;
  const float* x = (const float*)d_in[0];
  const float* Wm = (const float*)d_in[1];
  float* out = (float*)d_out;

  char* ws = (char*)d_ws;
  float* sq     = (float*)(ws + 0);        // 32768 f
  float* hsum   = (float*)(ws + 131072);   // 16 f
  float* inv2h2 = (float*)(ws + 131136);   // 16 f
  float* Ptab   = (float*)(ws + 131200);   // 2049 f
  float* rho    = (float*)(ws + 139520);   // 32768 f
  int*   rnk    = (int*)  (ws + 270592);   // 32768 i  (total < 0.4 MB)

  sq_kernel<<<Bb * Mm / 8, 256, 0, stream>>>(x, sq, hsum);
  gram_kernel<<<dim3(Mm / 128, Bb), 256, 0, stream>>>(x, sq, nullptr, hsum,
                                                      nullptr, 0);
  finalize_kernel<<<1, 256, 0, stream>>>(hsum, inv2h2, Ptab);
  gram_kernel<<<dim3(Mm / 128, Bb), 256, 0, stream>>>(x, sq, inv2h2, nullptr,
                                                      rho, 1);
  rank_kernel<<<Bb, 256, 0, stream>>>(rho, rnk);
  prop_kernel<<<dim3(Mm / 128, Bb), 256, 0, stream>>>(x, rnk, Ptab, out);
  outproj_kernel<<<Bb * Mm / 128, 256, 0, stream>>>(Wm, out);
}